// GCN_81329500717148
// MI455X (gfx1250) — compile-verified
//
#include <hip/hip_runtime.h>
#include <hip/hip_bf16.h>
#include <cstddef>
#include <cstdint>

// ---------------------------------------------------------------------------
// GCN forward on MI455X (gfx1250, wave32, WMMA).
//   h1 = elu(spmm(A, x@W1) + b1)
//   z  = elu(spmm(A, h1@W2) + b2)
//   h3 = elu(z@fcW1 + fcb1)
//   out= h3@fcW2 + fcb2
// Dense GEMMs: V_WMMA_F32_16X16X4_F32 (exact f32, matrix pipe), A tile staged
// into LDS via GLOBAL_LOAD_ASYNC_TO_LDS_B128 (ASYNCcnt), two 16-row tiles per
// wave sharing one B fragment (1 B-load per WMMA).
// SpMM: wave-per-edge float4 gather (support is L2-resident, 51 MB < 192 MB)
// + hardware global_atomic_add_f32 scatter.
// ---------------------------------------------------------------------------

typedef __attribute__((ext_vector_type(2))) float v2f;
typedef __attribute__((ext_vector_type(8))) float v8f;

#define N_NODES 100000
#define N_EDGES 1600000
#define NHID    128

__device__ __forceinline__ float elu_f(float v) {
    return v > 0.0f ? v : (expf(v) - 1.0f);
}

// Async copy 16 bytes global -> LDS (no VGPR round trip, ASYNCcnt tracked).
__device__ __forceinline__ void async_copy_b128(uint32_t lds_byte_off,
                                                const float* gaddr) {
    asm volatile("global_load_async_to_lds_b128 %0, %1, off"
                 :: "v"(lds_byte_off), "v"(gaddr)
                 : "memory");
}
__device__ __forceinline__ void wait_async_all() {
    asm volatile("s_wait_asynccnt 0x0" ::: "memory");
}

// ---------------------------------------------------------------------------
// init: acc[i*NHID + j] = bias[j]
// ---------------------------------------------------------------------------
__global__ __launch_bounds__(256) void init_bias_kernel(
    float* __restrict__ acc, const float* __restrict__ bias, int total) {
    int idx = blockIdx.x * blockDim.x + threadIdx.x;
    if (idx < total) acc[idx] = bias[idx & (NHID - 1)];
}

// ---------------------------------------------------------------------------
// elu in place
// ---------------------------------------------------------------------------
__global__ __launch_bounds__(256) void elu_kernel(float* __restrict__ v, int total) {
    int idx = blockIdx.x * blockDim.x + threadIdx.x;
    if (idx < total) v[idx] = elu_f(v[idx]);
}

// ---------------------------------------------------------------------------
// SpMM scatter: acc[rows[e], :] += vals[e] * S[cols[e], :]   (NH = 128)
// One wave per edge: 32 lanes x float4 = 128 features.
// ---------------------------------------------------------------------------
__global__ __launch_bounds__(256) void spmm_scatter_kernel(
    const int* __restrict__ rows, const int* __restrict__ cols,
    const float* __restrict__ vals, const float* __restrict__ S,
    float* __restrict__ acc, int nEdges) {
    const int wave = threadIdx.x >> 5;
    const int lane = threadIdx.x & 31;
    const int e = blockIdx.x * (256 / 32) + wave;
    if (e >= nEdges) return;

    const int r = rows[e];
    const int c = cols[e];
    const float v = vals[e];

    const float4 s = *reinterpret_cast<const float4*>(S + (size_t)c * NHID + lane * 4);
    float* dst = acc + (size_t)r * NHID + lane * 4;
    unsafeAtomicAdd(dst + 0, v * s.x);
    unsafeAtomicAdd(dst + 1, v * s.y);
    unsafeAtomicAdd(dst + 2, v * s.z);
    unsafeAtomicAdd(dst + 3, v * s.w);
}

// ---------------------------------------------------------------------------
// WMMA f32 GEMM: out[M,N] = X[M,K] @ W[K,N] (+bias, +elu per EPI)
//   EPI = 0: none, 1: bias+elu, 2: bias
// Workgroup (8 wave32) owns a 32-row block. A tile (32 x K) staged into LDS
// asynchronously; +4-float row pad keeps b128 stores 16-B aligned and makes
// the ds_load_b64 fragment reads bank-conflict-free (row stride mod 256B is
// 16/48B). Each wave owns a 16-col tile and both 16-row halves -> one B
// fragment feeds two WMMAs.
//
// Fragment layouts (ISA 7.12.2, wave32):
//   A 16x4 : lane l -> M = l&15, K = {kb, kb+1}, kb = (l>>4)*2   (v2f)
//   B 4x16 : lane l -> N = l&15, K = {kb, kb+1}                  (v2f)
//   C 16x16: vgpr r -> M = r + (l>=16 ? 8 : 0), N = l&15         (v8f)
// ---------------------------------------------------------------------------
template <int K, int N, int EPI>
__global__ __launch_bounds__(256) void wmma_gemm_kernel(
    const float* __restrict__ X, const float* __restrict__ W,
    const float* __restrict__ bias, float* __restrict__ out, int M) {
    constexpr int LDK = K + 4;               // padded row (floats)
    __shared__ float ldsA[32][LDK];

    const int m0  = blockIdx.x * 32;
    const int tid = threadIdx.x;

    // Async staged load of the 32xK A tile (K % 4 == 0, rows never straddled).
    {
        const uint32_t ldsBase = (uint32_t)(uintptr_t)(&ldsA[0][0]);
        for (int idx = tid * 4; idx < 32 * K; idx += 256 * 4) {
            const int r = idx / K;
            const int c = idx % K;           // multiple of 4, c+3 < K
            async_copy_b128(ldsBase + (uint32_t)(r * LDK + c) * 4u,
                            X + (size_t)(m0 + r) * K + c);
        }
        wait_async_all();
    }
    __syncthreads();

    const int wave = tid >> 5;
    const int lane = tid & 31;
    const int mrow = lane & 15;
    const int kb   = (lane >> 4) << 1;       // 0 or 2
    const int NT   = (N + 15) / 16;

    for (int t = wave; t < NT; t += 8) {
        const int n0 = t * 16;
        const int n  = n0 + (lane & 15);
        const bool nok = (n < N);

        v8f acc0 = {0.f, 0.f, 0.f, 0.f, 0.f, 0.f, 0.f, 0.f};
        v8f acc1 = {0.f, 0.f, 0.f, 0.f, 0.f, 0.f, 0.f, 0.f};

        for (int k = 0; k < K; k += 4) {
            const v2f a0 = *reinterpret_cast<const v2f*>(&ldsA[mrow][k + kb]);
            const v2f a1 = *reinterpret_cast<const v2f*>(&ldsA[mrow + 16][k + kb]);
            v2f b;
            b.x = nok ? W[(size_t)(k + kb + 0) * N + n] : 0.0f;
            b.y = nok ? W[(size_t)(k + kb + 1) * N + n] : 0.0f;
            acc0 = __builtin_amdgcn_wmma_f32_16x16x4_f32(
                false, a0, false, b, (short)0, acc0, false, false);
            acc1 = __builtin_amdgcn_wmma_f32_16x16x4_f32(
                false, a1, false, b, (short)0, acc1, false, false);
        }

        if (nok) {
            const int mbase = m0 + ((lane >> 4) << 3);
            float bv = 0.0f;
            if constexpr (EPI != 0) bv = bias[n];
#pragma unroll
            for (int r = 0; r < 8; ++r) {
                float v0 = acc0[r];
                float v1 = acc1[r];
                if constexpr (EPI != 0) { v0 += bv; v1 += bv; }   // no +0 in EPI=0 path
                if constexpr (EPI == 1) { v0 = elu_f(v0); v1 = elu_f(v1); }
                out[(size_t)(mbase + r) * N + n]      = v0;
                out[(size_t)(mbase + 16 + r) * N + n] = v1;
            }
        }
    }
}

// ---------------------------------------------------------------------------
// Host-side launcher
// ---------------------------------------------------------------------------
extern "C" void kernel_launch(void* const* d_in, const int* in_sizes, int n_in,
                              void* d_out, int out_size, void* d_ws, size_t ws_size,
                              hipStream_t stream) {
    const float* x    = (const float*)d_in[0];
    const float* W1   = (const float*)d_in[1];
    const float* b1   = (const float*)d_in[2];
    const float* W2   = (const float*)d_in[3];
    const float* b2   = (const float*)d_in[4];
    const float* fcW1 = (const float*)d_in[5];
    const float* fcb1 = (const float*)d_in[6];
    const float* fcW2 = (const float*)d_in[7];
    const float* fcb2 = (const float*)d_in[8];
    const int*   rows = (const int*)d_in[9];
    const int*   cols = (const int*)d_in[10];
    const float* vals = (const float*)d_in[11];
    float* out = (float*)d_out;

    // Workspace layout (floats):
    //   support : [100000,128]  12.8M   (reused for layer 2's support)
    //   h1      : [100000,128]  12.8M   (accum1 -> elu)
    //   z       : [100000,128]  12.8M   (accum2 -> elu)
    //   h3      : [100000,200]  20.0M
    float* ws      = (float*)d_ws;
    float* support = ws;
    float* h1      = ws + (size_t)12800000;
    float* z       = ws + (size_t)25600000;
    float* h3      = ws + (size_t)38400000;

    const int M = N_NODES;
    const int hidTotal = M * NHID;                 // 12.8M
    const int gElem    = (hidTotal + 255) / 256;   // elementwise grid
    const int gGemm    = M / 32;                   // 3125 row blocks (exact)
    const int gSpmm    = (N_EDGES + 7) / 8;        // wave-per-edge, 8 waves/block

    // ---- Layer 1: h1 = elu(spmm(x@W1) + b1)
    init_bias_kernel<<<gElem, 256, 0, stream>>>(h1, b1, hidTotal);
    wmma_gemm_kernel<256, 128, 0><<<gGemm, 256, 0, stream>>>(x, W1, nullptr, support, M);
    spmm_scatter_kernel<<<gSpmm, 256, 0, stream>>>(rows, cols, vals, support, h1, N_EDGES);
    elu_kernel<<<gElem, 256, 0, stream>>>(h1, hidTotal);

    // ---- Layer 2: z = elu(spmm(h1@W2) + b2)
    init_bias_kernel<<<gElem, 256, 0, stream>>>(z, b2, hidTotal);
    wmma_gemm_kernel<128, 128, 0><<<gGemm, 256, 0, stream>>>(h1, W2, nullptr, support, M);
    spmm_scatter_kernel<<<gSpmm, 256, 0, stream>>>(rows, cols, vals, support, z, N_EDGES);
    elu_kernel<<<gElem, 256, 0, stream>>>(z, hidTotal);

    // ---- MLP head
    wmma_gemm_kernel<128, 200, 1><<<gGemm, 256, 0, stream>>>(z, fcW1, fcb1, h3, M);
    wmma_gemm_kernel<200, 40, 2><<<gGemm, 256, 0, stream>>>(h3, fcW2, fcb2, out, M);
}